// PatchGenerator_81776177315975
// MI455X (gfx1250) — compile-verified
//
#include <hip/hip_runtime.h>
#include <math.h>
#include <stdint.h>

// ---------------- model constants ----------------
constexpr int   G       = 100;        // patch centers (10x10 grid)
constexpr int   GPAD    = 112;        // padded to 7 WMMA column tiles
constexpr int   P       = 32;         // MAX_PATCHES
constexpr int   GXY     = 80;
constexpr int   GZ      = 32;
constexpr int   V       = GXY * GXY * GZ;   // 204800 voxels / patch
constexpr int   MAX_VOX = 5000;
constexpr int   MAX_PPV = 5;
constexpr int   NF      = 5;
constexpr int   MIN_PTS = 30;
constexpr float RAD     = 10.0f;
constexpr float VOX     = 0.25f;
constexpr float LOX = -75.2f, LOY = -75.2f, LOZ = -5.0f;
constexpr float HIX =  75.2f, HIY =  75.2f, HIZ =  3.0f;
constexpr int   CHUNK  = 1024;
constexpr int   NCHUNK = V / CHUNK;   // 200
constexpr int   PTS_PER_BLK = 128;    // 8 waves x 16-point WMMA tiles

// ---- output layout (4-byte elements, flat concat in return order) ----
constexpr long long N_FEATE   = (long long)P * MAX_VOX * MAX_PPV * NF; // 4,000,000
constexpr long long OFF_COORD = N_FEATE;
constexpr long long N_COORD   = (long long)P * MAX_VOX * 4;            // 640,000
constexpr long long OFF_NPTS  = OFF_COORD + N_COORD;
constexpr long long N_NPTS    = (long long)P * MAX_VOX;                // 160,000
constexpr long long OFF_SELC  = OFF_NPTS + N_NPTS;
constexpr long long OFF_SELO  = OFF_SELC + 2 * P;

typedef __attribute__((ext_vector_type(2)))  float    v2f;
typedef __attribute__((ext_vector_type(8)))  float    v8f;

// async-to-LDS builtin parameter types (from hipcc diagnostic: AS1/AS3 int4*)
typedef int v4i __attribute__((__vector_size__(4 * sizeof(int))));
typedef __attribute__((address_space(1))) v4i as1_v4i;
typedef __attribute__((address_space(3))) v4i as3_v4i;

#if __has_builtin(__builtin_amdgcn_global_load_async_to_lds_b128)
#define HAVE_ASYNC_LDS 1
#endif

__device__ __forceinline__ float centerX(int g) { return ((g % 10) + 0.5f) * 15.0f - 75.2f; }
__device__ __forceinline__ float centerY(int g) { return ((g / 10) + 0.5f) * 15.0f - 75.2f; }

__device__ __forceinline__ bool inRange(float x, float y, float z) {
    return x > LOX && x < HIX - 1e-3f &&
           y > LOY && y < HIY - 1e-3f &&
           z > LOZ && z < HIZ - 1e-3f;
}

__device__ __forceinline__ void wait_async_lds() {
#if defined(HAVE_ASYNC_LDS)
#if __has_builtin(__builtin_amdgcn_s_wait_asynccnt)
    __builtin_amdgcn_s_wait_asynccnt(0);
#else
    asm volatile("s_wait_asynccnt 0x0" ::: "memory");
#endif
#endif
}

// ---------------- init kernels ----------------
__global__ void zero_f32(float* p, long long n) {
    for (long long i = blockIdx.x * (long long)blockDim.x + threadIdx.x;
         i < n; i += (long long)gridDim.x * blockDim.x) p[i] = 0.0f;
}
__global__ void zero_i32(int* p, long long n) {
    for (long long i = blockIdx.x * (long long)blockDim.x + threadIdx.x;
         i < n; i += (long long)gridDim.x * blockDim.x) p[i] = 0;
}
__global__ void init_coord_npts(int* coord, int* npts) {
    long long i = blockIdx.x * (long long)blockDim.x + threadIdx.x;
    if (i >= (long long)P * MAX_VOX) return;
    coord[i * 4 + 0] = (int)(i / MAX_VOX);  // patch id present for every row
    coord[i * 4 + 1] = 0;
    coord[i * 4 + 2] = 0;
    coord[i * 4 + 3] = 0;
    npts[i] = 0;
}

// ---------------- stage 1: WMMA membership counting ----------------
// Block stages 128 points (2560 B) into LDS via async-to-LDS (ASYNCcnt), then
// 8 waves each run 7x V_WMMA_F32_16X16X4_F32:
//   A row m = [x, y, x^2+y^2, 1]        B col n = [-2cx, -2cy, 1, cx^2+cy^2]
//   => acc = d^2(point m, center n)
// A 16x4 f32 layout: lanes 0-15 hold K0,K1; lanes 16-31 hold K2,K3 (ISA 7.12.2).
// Out-of-range / tail points poison K2 with 1e30 so d^2 fails the (< R^2) test.
__global__ void __launch_bounds__(256) count_members(const float* __restrict__ pts,
                                                     int n, int* __restrict__ gcnt) {
    __shared__ int lcnt[G];
    __shared__ __align__(16) float spts[PTS_PER_BLK * 5];
    for (int i = threadIdx.x; i < G; i += blockDim.x) lcnt[i] = 0;

    const int base = blockIdx.x * PTS_PER_BLK;       // always < n (grid sizing)
    const int npb  = min(PTS_PER_BLK, n - base);
    const int ndw  = npb * 5;
    if (ndw == PTS_PER_BLK * 5) {
#if defined(HAVE_ASYNC_LDS)
        for (int q = threadIdx.x; q < ndw / 4; q += blockDim.x)
            __builtin_amdgcn_global_load_async_to_lds_b128(
                (as1_v4i*)(uintptr_t)(pts + (size_t)base * 5 + q * 4),
                (as3_v4i*)(uintptr_t)(spts + q * 4), 0, 0);
#else
        for (int d = threadIdx.x; d < ndw; d += blockDim.x)
            spts[d] = pts[(size_t)base * 5 + d];
#endif
    } else {                                          // tail block: plain copy
        for (int d = threadIdx.x; d < ndw; d += blockDim.x)
            spts[d] = pts[(size_t)base * 5 + d];
    }
    wait_async_lds();
    __syncthreads();

    const int wave = threadIdx.x >> 5;
    const int lane = threadIdx.x & 31;
    const int m    = lane & 15;

    if (base + wave * 16 < n) {                      // wave-uniform guard
        const int pt = base + wave * 16 + m;
        float x = 0.0f, y = 0.0f, z = 0.0f;
        if (pt < n) {
            const int l = (wave * 16 + m) * 5;
            x = spts[l]; y = spts[l + 1]; z = spts[l + 2];
        }
        const bool ok = (pt < n) && inRange(x, y, z);

        v2f a;
        if (lane < 16) { a[0] = x; a[1] = y; }
        else           { a[0] = ok ? (x * x + y * y) : 1e30f; a[1] = 1.0f; }

        for (int ct = 0; ct < 7; ++ct) {
            const int  cidx = ct * 16 + m;
            const float cx = centerX(cidx), cy = centerY(cidx);
            v2f b;
            if (lane < 16) { b[0] = -2.0f * cx; b[1] = -2.0f * cy; }
            else           { b[0] = 1.0f;       b[1] = cx * cx + cy * cy; }

            v8f acc = {};
            acc = __builtin_amdgcn_wmma_f32_16x16x4_f32(
                false, a, false, b, (short)0, acc, false, false);

            // column n = lane&15 lives in this lane across 8 accumulator regs
            int local = 0;
#pragma unroll
            for (int r = 0; r < 8; ++r)
                if (acc[r] < RAD * RAD) ++local;
            if (cidx < G && local) atomicAdd(&lcnt[cidx], local);
        }
    }
    __syncthreads();
    for (int i = threadIdx.x; i < G; i += blockDim.x)
        if (lcnt[i]) atomicAdd(&gcnt[i], lcnt[i]);
}

// ---------------- stage 2: top-32 patch selection ----------------
__global__ void select_patches(const int* __restrict__ gcnt,
                               float* selc, float* selo, int* idx, int* validf) {
    if (threadIdx.x != 0 || blockIdx.x != 0) return;
    float score[G]; int used[G];
    for (int g = 0; g < G; ++g) {
        float cx = centerX(g), cy = centerY(g);
        score[g] = (gcnt[g] >= MIN_PTS) ? sqrtf(cx * cx + cy * cy) : __builtin_inff();
        used[g]  = 0;
    }
    for (int k = 0; k < P; ++k) {
        int best = -1; float bs = __builtin_inff();
        for (int g = 0; g < G; ++g) {
            if (used[g]) continue;
            if (best < 0 || score[g] < bs) { best = g; bs = score[g]; } // tie -> lowest idx
        }
        used[best] = 1;
        float cx = centerX(best), cy = centerY(best);
        selc[2 * k] = cx; selc[2 * k + 1] = cy;
        selo[k] = atan2f(cy, cx);
        idx[k] = best;
        validf[k] = (gcnt[best] >= MIN_PTS) ? 1 : 0;
    }
}

// ---------------- per-point patch loop kernels ----------------
// One thread per point; 32 patch params staged in LDS. Point data read once.
struct PatchLds { float cx[P], cy[P], co[P], si[P]; int val[P]; };

__device__ __forceinline__ void load_patches(PatchLds& pl, const float* selc,
                                             const float* selo, const int* validf) {
    if (threadIdx.x < P) {
        int p = threadIdx.x;
        pl.cx[p] = selc[2 * p]; pl.cy[p] = selc[2 * p + 1];
        float o = selo[p];
        pl.co[p] = cosf(o); pl.si[p] = sinf(o);
        pl.val[p] = validf[p];
    }
    __syncthreads();
}

__device__ __forceinline__ bool voxel_of(float dx, float dy, float z,
                                         float c, float s,
                                         float& px, float& py, int& vid) {
    if (dx * dx + dy * dy >= RAD * RAD) return false;
    px = dx * c + dy * s;                 // [dx,dy] @ [[c,-s],[s,c]]
    py = -dx * s + dy * c;
    int gx = (int)floorf((px + RAD) / VOX);
    int gy = (int)floorf((py + RAD) / VOX);
    int gz = (int)floorf((z + 5.0f) / VOX);
    if (gx < 0 || gx >= GXY || gy < 0 || gy >= GXY || gz < 0 || gz >= GZ) return false;
    vid = gz * (GXY * GXY) + gy * GXY + gx;
    return true;
}

__global__ void vox_count(const float* __restrict__ pts, int n,
                          const float* __restrict__ selc, const float* __restrict__ selo,
                          const int* __restrict__ validf, int* __restrict__ vox) {
    __shared__ PatchLds pl;
    load_patches(pl, selc, selo, validf);
    int pt = blockIdx.x * blockDim.x + threadIdx.x;
    if (pt >= n) return;
    __builtin_prefetch(pts + (size_t)pt * 5, 0, 3);
    float x = pts[pt * 5], y = pts[pt * 5 + 1], z = pts[pt * 5 + 2];
    if (!inRange(x, y, z)) return;
    for (int p = 0; p < P; ++p) {
        if (!pl.val[p]) continue;
        float px, py; int vid;
        if (voxel_of(x - pl.cx[p], y - pl.cy[p], z, pl.co[p], pl.si[p], px, py, vid))
            atomicAdd(&vox[(long long)p * V + vid], 1);
    }
}

__global__ void scatter_feat(const float* __restrict__ pts, int n,
                             const float* __restrict__ selc, const float* __restrict__ selo,
                             const int* __restrict__ validf, const int* __restrict__ vox,
                             float* __restrict__ feat, int* __restrict__ npts) {
    __shared__ PatchLds pl;
    load_patches(pl, selc, selo, validf);
    int pt = blockIdx.x * blockDim.x + threadIdx.x;
    if (pt >= n) return;
    float x = pts[pt * 5], y = pts[pt * 5 + 1], z = pts[pt * 5 + 2];
    float e0 = pts[pt * 5 + 3], e1 = pts[pt * 5 + 4];
    if (!inRange(x, y, z)) return;
    for (int p = 0; p < P; ++p) {
        if (!pl.val[p]) continue;
        float px, py; int vid;
        if (!voxel_of(x - pl.cx[p], y - pl.cy[p], z, pl.co[p], pl.si[p], px, py, vid)) continue;
        int s = vox[(long long)p * V + vid];
        if (s < 0) continue;
        long long row = (long long)p * MAX_VOX + s;
        int r = atomicAdd(&npts[row], 1);
        if (r < MAX_PPV) {
            float* f = feat + (row * MAX_PPV + r) * NF;
            f[0] = px; f[1] = py; f[2] = z; f[3] = e0; f[4] = e1;
        }
    }
}

// ---------------- stage 4: compaction scan (per patch) ----------------
__global__ void scan_partial(const int* __restrict__ vox, int* __restrict__ part) {
    int c = blockIdx.x * blockDim.x + threadIdx.x;   // chunk id
    if (c >= P * NCHUNK) return;
    long long base = (long long)c * CHUNK;
    int s = 0;
    for (int i = 0; i < CHUNK; ++i) s += (vox[base + i] > 0);
    part[c] = s;
}
__global__ void scan_exclusive(int* part) {
    int p = threadIdx.x;
    if (p >= P) return;
    int run = 0;
    for (int i = 0; i < NCHUNK; ++i) {
        int v = part[p * NCHUNK + i];
        part[p * NCHUNK + i] = run;
        run += v;
    }
}
__global__ void scan_apply(int* __restrict__ vox, const int* __restrict__ part) {
    int c = blockIdx.x * blockDim.x + threadIdx.x;
    if (c >= P * NCHUNK) return;
    long long base = (long long)c * CHUNK;
    int run = part[c];                               // patch-relative slot base
    for (int i = 0; i < CHUNK; ++i) {
        int cnt = vox[base + i];
        vox[base + i] = (cnt > 0 && run < MAX_VOX) ? run : -1;
        run += (cnt > 0);
    }
}

// ---------------- stage 5: voxel coords ----------------
__global__ void write_coords(const int* __restrict__ vox, int* __restrict__ coord) {
    long long i = blockIdx.x * (long long)blockDim.x + threadIdx.x;
    if (i >= (long long)P * V) return;
    int s = vox[i];
    if (s < 0) return;
    int p = (int)(i / V), v = (int)(i % V);
    long long row = (long long)p * MAX_VOX + s;
    coord[row * 4 + 0] = p;
    coord[row * 4 + 1] = v / (GXY * GXY);
    coord[row * 4 + 2] = (v / GXY) % GXY;
    coord[row * 4 + 3] = v % GXY;
}

__global__ void clamp_npts(int* npts) {
    long long i = blockIdx.x * (long long)blockDim.x + threadIdx.x;
    if (i >= (long long)P * MAX_VOX) return;
    int v = npts[i];
    npts[i] = v > MAX_PPV ? MAX_PPV : v;
}

// ---------------- launch ----------------
extern "C" void kernel_launch(void* const* d_in, const int* in_sizes, int n_in,
                              void* d_out, int out_size, void* d_ws, size_t ws_size,
                              hipStream_t stream) {
    (void)n_in; (void)out_size; (void)ws_size;
    const float* pts = (const float*)d_in[0];
    const int n = in_sizes[0] / 5;

    float* out   = (float*)d_out;
    float* feat  = out;
    int*   coord = (int*)(out + OFF_COORD);
    int*   npts  = (int*)(out + OFF_NPTS);
    float* selc  = out + OFF_SELC;
    float* selo  = out + OFF_SELO;

    char* ws    = (char*)d_ws;
    int* vox    = (int*)ws;                                       // P*V ints (26.2 MB)
    int* part   = (int*)(ws + (size_t)P * V * 4);                 // P*NCHUNK ints
    int* gcnt   = part + P * NCHUNK;                              // GPAD ints
    int* idx    = gcnt + GPAD;                                    // P ints
    int* validf = idx + P;                                        // P ints

    // deterministic per-call init (harness does not re-poison between replays)
    zero_f32<<<1024, 256, 0, stream>>>(feat, N_FEATE);
    zero_i32<<<1024, 256, 0, stream>>>(vox, (long long)P * V);
    zero_i32<<<1, 256, 0, stream>>>(gcnt, GPAD);
    init_coord_npts<<<(P * MAX_VOX + 255) / 256, 256, 0, stream>>>(coord, npts);

    count_members<<<(n + PTS_PER_BLK - 1) / PTS_PER_BLK, 256, 0, stream>>>(pts, n, gcnt);
    select_patches<<<1, 32, 0, stream>>>(gcnt, selc, selo, idx, validf);

    const int pblk = (n + 255) / 256;
    vox_count<<<pblk, 256, 0, stream>>>(pts, n, selc, selo, validf, vox);

    scan_partial<<<(P * NCHUNK + 255) / 256, 256, 0, stream>>>(vox, part);
    scan_exclusive<<<1, 32, 0, stream>>>(part);
    scan_apply<<<(P * NCHUNK + 255) / 256, 256, 0, stream>>>(vox, part);

    write_coords<<<(int)(((long long)P * V + 255) / 256), 256, 0, stream>>>(vox, coord);
    scatter_feat<<<pblk, 256, 0, stream>>>(pts, n, selc, selo, validf, vox, feat, npts);
    clamp_npts<<<(P * MAX_VOX + 255) / 256, 256, 0, stream>>>(npts);
}